// TTCLoss_53231824667083
// MI455X (gfx1250) — compile-verified
//
#include <hip/hip_runtime.h>
#include <math.h>

typedef __attribute__((ext_vector_type(2))) float v2f;
typedef __attribute__((ext_vector_type(4))) float v4f;
typedef __attribute__((ext_vector_type(8))) float v8f;

#define NB 1024
#define NT 256
#define NWAVES (NT / 32)

// ---------------------------------------------------------------------------
// Exact-f32 wave reduction via V_WMMA_F32_16X16X4_F32 with B = ones.
// A[m][0] = acc(lane m), A[m][2] = acc(lane 16+m), other K slots = 0.
// D[m][n] = acc_m + acc_{m+16}. Lane L<16 holds rows 0..7 in its 8 D VGPRs,
// lanes 16..31 hold rows 8..15, so the per-lane sum of the 8 D VGPRs gives
// the half-wave totals at lanes 0 and 16.
// ---------------------------------------------------------------------------
__device__ __forceinline__ float wmma_wave_sum(float acc) {
  v2f a; a[0] = acc;  a[1] = 0.0f;
  v2f b; b[0] = 1.0f; b[1] = 1.0f;
  v8f c = {0.0f, 0.0f, 0.0f, 0.0f, 0.0f, 0.0f, 0.0f, 0.0f};
  c = __builtin_amdgcn_wmma_f32_16x16x4_f32(false, a, false, b, (short)0, c,
                                            false, false);
  return c[0] + c[1] + c[2] + c[3] + c[4] + c[5] + c[6] + c[7];
}

// Block-level sum; result valid at tid==0. EXEC must be all-ones on entry.
__device__ __forceinline__ float block_sum_wmma(float acc, float* smem) {
  const int tid  = threadIdx.x;
  const int lane = tid & 31;
  const int wave = tid >> 5;
  float wsum = wmma_wave_sum(acc);
  if (lane == 0)  smem[2 * wave + 0] = wsum;  // rows 0..7  of this wave
  if (lane == 16) smem[2 * wave + 1] = wsum;  // rows 8..15 of this wave
  __syncthreads();
  float tot = 0.0f;
  if (tid == 0) {
#pragma unroll
    for (int i = 0; i < 2 * NWAVES; ++i) tot += smem[i];
  }
  return tot;
}

// Online softmax-stat merge: (m1,s1) <- combine((m1,s1),(m2,s2))
__device__ __forceinline__ void online_combine(float& m1, float& s1,
                                               float m2, float s2) {
  float M  = fmaxf(m1, m2);
  float e1 = __expf(m1 - M);   // exp(-inf) == 0 handles identity partials
  float e2 = __expf(m2 - M);
  s1 = s1 * e1 + s2 * e2;
  m1 = M;
}

__device__ __forceinline__ float check_values(float x) {
  // if (0 <= x <= 0.5) or isnan(x): x = 0.6
  bool bad = ((x >= 0.0f) && (x <= 0.5f)) || (x != x);
  return bad ? 0.6f : x;
}

// Full per-element loss term given softmax numerator exp(p - M) * invS.
__device__ __forceinline__ float loss_term(float p_raw, float t_raw, float M,
                                           float invS) {
  // preds: softmax -> recenter/rescale -> clip -> check_values
  float p = __expf(p_raw - M) * invS;
  p = (p - 0.5f) * 20.0f;
  p = fminf(fmaxf(p, -10.0f), 10.0f);
  p = check_values(p);
  float lp = __logf(1.0f - 0.5f / p);
  // targets: check_values -> clip
  float t = check_values(t_raw);
  t = fminf(fmaxf(t, -10.0f), 10.0f);
  float lt = __logf(1.0f - 0.5f / t);
  return fabsf(lp - lt);
}

// ---------------------------------------------------------------------------
// Kernel 1: per-block online (max, sum-exp) over preds. One read of preds,
// default RT temporal hint so preds stays resident in the 192 MB L2 for the
// loss pass.
// ---------------------------------------------------------------------------
__global__ void k_stats(const float* __restrict__ preds, int n,
                        float* __restrict__ ws_m, float* __restrict__ ws_s) {
  const int tid = threadIdx.x;
  const int bid = blockIdx.x;
  const long stride = (long)NB * NT;
  const long n4 = (long)n >> 2;
  const v4f* p4 = reinterpret_cast<const v4f*>(preds);

  float m = -INFINITY, s = 0.0f;
  for (long i = (long)bid * NT + tid; i < n4; i += stride) {
    v4f v = p4[i];
#pragma unroll
    for (int j = 0; j < 4; ++j) {
      float x = v[j];
      if (x > m) { s *= __expf(m - x); m = x; }  // rare rescale
      s += __expf(x - m);
    }
  }
  if (bid == 0) {  // scalar tail (empty for n = 2^25)
    for (long i = (n4 << 2) + tid; i < n; i += NT) {
      float x = preds[i];
      if (x > m) { s *= __expf(m - x); m = x; }
      s += __expf(x - m);
    }
  }

  __shared__ float sm[NT], ss[NT];
  sm[tid] = m; ss[tid] = s;
  __syncthreads();
  for (int off = NT / 2; off > 0; off >>= 1) {
    if (tid < off) {
      float m1 = sm[tid], s1 = ss[tid];
      online_combine(m1, s1, sm[tid + off], ss[tid + off]);
      sm[tid] = m1; ss[tid] = s1;
    }
    __syncthreads();
  }
  if (tid == 0) { ws_m[bid] = sm[0]; ws_s[bid] = ss[0]; }
}

// ---------------------------------------------------------------------------
// Kernel 2: combine NB (m,s) pairs -> global (M, S) in ws[0], ws[1].
// ---------------------------------------------------------------------------
__global__ void k_combine(const float* __restrict__ ws_m,
                          const float* __restrict__ ws_s,
                          float* __restrict__ ws_out) {
  const int tid = threadIdx.x;
  float m = -INFINITY, s = 0.0f;
  for (int i = tid; i < NB; i += NT) online_combine(m, s, ws_m[i], ws_s[i]);

  __shared__ float sm[NT], ss[NT];
  sm[tid] = m; ss[tid] = s;
  __syncthreads();
  for (int off = NT / 2; off > 0; off >>= 1) {
    if (tid < off) {
      float m1 = sm[tid], s1 = ss[tid];
      online_combine(m1, s1, sm[tid + off], ss[tid + off]);
      sm[tid] = m1; ss[tid] = s1;
    }
    __syncthreads();
  }
  if (tid == 0) { ws_out[0] = sm[0]; ws_out[1] = ss[0]; }
}

// ---------------------------------------------------------------------------
// Kernel 3: loss partials. Both streams loaded non-temporally
// (global_load_b128 th:NT): preds still HITS L2 (resident from kernel 1) but
// is marked for early eviction (last use), and the single-use targets stream
// does not displace preds lines mid-kernel. Unroll x2 keeps 4 B128 loads in
// flight per loop iteration.
// ---------------------------------------------------------------------------
__global__ void k_loss(const float* __restrict__ preds,
                       const float* __restrict__ tgts, int n,
                       const float* __restrict__ MS,
                       float* __restrict__ ws_l) {
  const int tid = threadIdx.x;
  const int bid = blockIdx.x;
  const float M    = MS[0];
  const float invS = 1.0f / MS[1];
  const long stride = (long)NB * NT;
  const long n4 = (long)n >> 2;
  const v4f* p4 = reinterpret_cast<const v4f*>(preds);
  const v4f* t4 = reinterpret_cast<const v4f*>(tgts);

  float acc = 0.0f;
  long i = (long)bid * NT + tid;
  // Unrolled-by-2 main loop (two independent B128 load pairs in flight).
  for (; i + stride < n4; i += 2 * stride) {
    v4f pv0 = __builtin_nontemporal_load(&p4[i]);
    v4f tv0 = __builtin_nontemporal_load(&t4[i]);
    v4f pv1 = __builtin_nontemporal_load(&p4[i + stride]);
    v4f tv1 = __builtin_nontemporal_load(&t4[i + stride]);
#pragma unroll
    for (int j = 0; j < 4; ++j) acc += loss_term(pv0[j], tv0[j], M, invS);
#pragma unroll
    for (int j = 0; j < 4; ++j) acc += loss_term(pv1[j], tv1[j], M, invS);
  }
  for (; i < n4; i += stride) {
    v4f pv = __builtin_nontemporal_load(&p4[i]);
    v4f tv = __builtin_nontemporal_load(&t4[i]);
#pragma unroll
    for (int j = 0; j < 4; ++j) acc += loss_term(pv[j], tv[j], M, invS);
  }
  if (bid == 0) {  // scalar tail (empty for n = 2^25)
    for (long k = (n4 << 2) + tid; k < n; k += NT)
      acc += loss_term(preds[k], tgts[k], M, invS);
  }

  __shared__ float smem[2 * NWAVES];
  float tot = block_sum_wmma(acc, smem);
  if (tid == 0) ws_l[bid] = tot;
}

// ---------------------------------------------------------------------------
// Kernel 4: sum NB loss partials, divide by avg_factor, write scalar output.
// ---------------------------------------------------------------------------
__global__ void k_final(const float* __restrict__ ws_l,
                        const float* __restrict__ avg,
                        float* __restrict__ out) {
  const int tid = threadIdx.x;
  float acc = 0.0f;
  for (int i = tid; i < NB; i += NT) acc += ws_l[i];
  __shared__ float smem[2 * NWAVES];
  float tot = block_sum_wmma(acc, smem);
  if (tid == 0) out[0] = tot / avg[0];
}

extern "C" void kernel_launch(void* const* d_in, const int* in_sizes, int n_in,
                              void* d_out, int out_size, void* d_ws,
                              size_t ws_size, hipStream_t stream) {
  (void)n_in; (void)out_size; (void)ws_size;
  const float* preds = (const float*)d_in[0];
  const float* tgts  = (const float*)d_in[1];
  const float* avg   = (const float*)d_in[2];
  float* out = (float*)d_out;
  float* ws  = (float*)d_ws;
  const int n = in_sizes[0];

  // ws layout (floats): [0]=M, [1]=S, [2..2+NB)=block max, then block sumexp,
  // then block loss partials. Every slot is written before it is read each
  // call, so no pre-initialization is needed (deterministic, atomics-free).
  float* ws_m = ws + 2;
  float* ws_s = ws_m + NB;
  float* ws_l = ws_s + NB;

  k_stats  <<<NB, NT, 0, stream>>>(preds, n, ws_m, ws_s);
  k_combine<<<1,  NT, 0, stream>>>(ws_m, ws_s, ws);
  k_loss   <<<NB, NT, 0, stream>>>(preds, tgts, n, ws, ws_l);
  k_final  <<<1,  NT, 0, stream>>>(ws_l, avg, out);
}